// Network_85615878078979
// MI455X (gfx1250) — compile-verified
//
#include <hip/hip_runtime.h>
#include <stdint.h>

#define IMG 32
#define NU 96
#define SHAPE (IMG * NU)            // 3072
#define NTILES (NU * NU)            // 9216
#define ELEMS ((size_t)SHAPE * SHAPE) // 9437184
#define EPS_LR 1e-7f

typedef float v2f __attribute__((ext_vector_type(2)));
typedef float v4f __attribute__((ext_vector_type(4)));
typedef float v8f __attribute__((ext_vector_type(8)));

__global__ void init_ws(unsigned long long* key) { *key = ~0ull; }

// One wave32 per 32x32 tile: z = sum((som - x)^2 / var).
// Wave reduction via V_WMMA_F32_16X16X4_F32 with B = ones (row-sum trick),
// then one cross-half shuffle. Block combines 8 waves in LDS -> 1 atomicMin_u64.
__global__ __launch_bounds__(256) void bmu_kernel(const float* __restrict__ som,
                                                  const float* __restrict__ var,
                                                  const float* __restrict__ x,
                                                  unsigned long long* __restrict__ gkey)
{
    const int tid  = threadIdx.x;
    const int lane = tid & 31;
    const int wid  = tid >> 5;
    const int tile = blockIdx.x * 8 + wid;          // 1152 blocks * 8 waves = 9216 tiles
    const int u = tile / NU, v = tile % NU;
    const size_t base = (size_t)u * (IMG * SHAPE) + (size_t)v * IMG;
    const int colf  = (lane & 7) * 4;               // float4 column within tile row
    const int rbase = lane >> 3;                    // 4 rows covered per iteration

    float acc = 0.f;
#pragma unroll
    for (int k = 0; k < 8; ++k) {
        const int r = k * 4 + rbase;
        const size_t off = base + (size_t)r * SHAPE + colf;
        v4f s  = *(const v4f*)(som + off);
        v4f w  = *(const v4f*)(var + off);
        v4f xv = *(const v4f*)(x + (r * IMG + colf));
        v4f d  = s - xv;
        acc += d.x * d.x * __builtin_amdgcn_rcpf(w.x);
        acc += d.y * d.y * __builtin_amdgcn_rcpf(w.y);
        acc += d.z * d.z * __builtin_amdgcn_rcpf(w.z);
        acc += d.w * d.w * __builtin_amdgcn_rcpf(w.w);
    }

    // A(16x4): VGPR0 = per-lane partial, VGPR1 = 0  ->  rowsum[m] = acc[m] + acc[m+16]
    // B(4x16) = ones  ->  D[m][j] = rowsum[m] for every column j.
    v2f a; a.x = acc; a.y = 0.f;
    v2f b; b.x = 1.f; b.y = 1.f;
    v8f c = {};
    c = __builtin_amdgcn_wmma_f32_16x16x4_f32(false, a, false, b,
                                              (short)0, c, false, false);
    float part = c[0] + c[1] + c[2] + c[3] + c[4] + c[5] + c[6] + c[7];
    // lanes 0-15 now hold sum(rowsum[0..7]), lanes 16-31 hold sum(rowsum[8..15])
    float z = part + __shfl_xor(part, 16);

    __shared__ unsigned long long skey[8];
    if (lane == 0)
        skey[wid] = ((unsigned long long)__float_as_uint(z) << 32) | (unsigned)tile;
    __syncthreads();
    if (tid == 0) {
        unsigned long long m = skey[0];
#pragma unroll
        for (int i = 1; i < 8; ++i) { unsigned long long s = skey[i]; m = (s < m) ? s : m; }
        atomicMin(gkey, m);
    }
}

// Per-tile neighborhood weights + tiny radius/lr outputs (9216 threads).
__global__ __launch_bounds__(256) void prep_kernel(const float* __restrict__ radius,
                                                   const float* __restrict__ lrates,
                                                   const float* __restrict__ bmu_count,
                                                   const unsigned long long* __restrict__ gkey,
                                                   float* __restrict__ fm_tab,
                                                   float* __restrict__ va_tab,
                                                   float* __restrict__ out_rad,
                                                   float* __restrict__ out_lr)
{
    const int idx = blockIdx.x * 256 + threadIdx.x;
    if (idx >= NTILES) return;
    const unsigned long long k = *gkey;
    const int tile = (int)(k & 0xffffffffull);
    const int bi = tile / NU, bj = tile % NU;

    const float r0  = radius[tile];
    const float lr0 = lrates[tile];
    const float dm  = 1.f / (2.f * r0 * r0);
    const float constant = -logf(EPS_LR / lr0) / dm;

    const int i = idx / NU, j = idx % NU;
    const float di = (float)i - (float)bi;
    const float dj = (float)j - (float)bj;
    const float cart = sqrtf(di * di + dj * dj);
    const float modifier = (cart > r0) ? 0.f : cart;

    fm_tab[idx] = lrates[idx] * expf(-modifier) * dm;
    const float sig = 1.f / (1.f + expf(-cart / constant));
    va_tab[idx] = fminf(fmaxf(0.4f + sig, 0.f), 1.f);   // ALPHA - 0.5 = 0.4

    const float b0 = bmu_count[tile * 10];
    const float nr = (idx == tile) ? expf(-b0 * (1.f / 15.f)) : radius[idx];
    const float nl = (idx == tile) ? expf(-b0 * (1.f / 25.f)) : lrates[idx];
    out_rad[idx] = fmaxf(nr, 1e-5f);
    out_lr[idx]  = fmaxf(nl, 1e-5f);
}

// Streaming update: one float4 per thread. som/var reads hit L2 (warmed by
// bmu_kernel); outputs use nontemporal stores to avoid evicting them.
__global__ __launch_bounds__(256) void update_kernel(const float* __restrict__ som,
                                                     const float* __restrict__ var,
                                                     const float* __restrict__ x,
                                                     const float* __restrict__ fm_tab,
                                                     const float* __restrict__ va_tab,
                                                     float* __restrict__ out_som,
                                                     float* __restrict__ out_var)
{
    const int p   = blockIdx.x * 256 + threadIdx.x;   // 0 .. 2359295 float4s
    const int row = p / 768;                          // 3072/4 float4s per row
    const int c4  = p - row * 768;
    const int t   = (row >> 5) * NU + (c4 >> 3);      // tile index
    const float fm = fm_tab[t];
    const float va = va_tab[t];

    const size_t off = (size_t)p * 4;
    v4f s  = *(const v4f*)(som + off);
    v4f w  = *(const v4f*)(var + off);
    v4f xv = *(const v4f*)(x + ((row & 31) * IMG + ((c4 & 7) << 2)));

    v4f ns = s + fm * (xv - s);          // unclipped, used for variance
    v4f d  = xv - ns;
    v4f nv = va * w + (1.f - va) * d * d;

    v4f zero = {0.f, 0.f, 0.f, 0.f};
    v4f one  = {1.f, 1.f, 1.f, 1.f};
    ns = __builtin_elementwise_min(__builtin_elementwise_max(ns, zero), one);

    __builtin_nontemporal_store(ns, (v4f*)(out_som + off));
    __builtin_nontemporal_store(nv, (v4f*)(out_var + off));
}

extern "C" void kernel_launch(void* const* d_in, const int* in_sizes, int n_in,
                              void* d_out, int out_size, void* d_ws, size_t ws_size,
                              hipStream_t stream) {
    (void)in_sizes; (void)n_in; (void)out_size; (void)ws_size;
    const float* x         = (const float*)d_in[0];
    const float* som       = (const float*)d_in[1];
    const float* var       = (const float*)d_in[2];
    const float* radius    = (const float*)d_in[3];
    const float* lrates    = (const float*)d_in[4];
    const float* bmu_count = (const float*)d_in[5];

    float* out     = (float*)d_out;
    float* out_som = out;
    float* out_var = out + ELEMS;
    float* out_rad = out + 2 * ELEMS;
    float* out_lr  = out_rad + NTILES;

    unsigned long long* gkey = (unsigned long long*)d_ws;
    float* fm_tab = (float*)((char*)d_ws + 16);
    float* va_tab = fm_tab + NTILES;

    init_ws<<<1, 1, 0, stream>>>(gkey);
    bmu_kernel<<<NTILES / 8, 256, 0, stream>>>(som, var, x, gkey);
    prep_kernel<<<(NTILES + 255) / 256, 256, 0, stream>>>(radius, lrates, bmu_count,
                                                          gkey, fm_tab, va_tab,
                                                          out_rad, out_lr);
    update_kernel<<<(int)(ELEMS / 4 / 256), 256, 0, stream>>>(som, var, x,
                                                              fm_tab, va_tab,
                                                              out_som, out_var);
}